// TabularEmbedding_38036230373568
// MI455X (gfx1250) — compile-verified
//
#include <hip/hip_runtime.h>
#include <hip/hip_bf16.h>
#include <stdint.h>

// Problem constants (compile-time, from the reference)
#define B_      2048
#define T_      13
#define D_      157
#define H_      8
#define NCAT    11
#define NCONT   (D_ - NCAT)           // 146
#define DH      (D_ * H_)             // 1256
#define NROWS   (B_ * T_)             // 26624
#define VOCABT  45
#define ROWS    16                    // rows per tile (26624 % 16 == 0)
#define TILE_F  (ROWS * D_)           // 2512 floats per tile
#define TILE_BYTES (TILE_F * 4)       // 10048 bytes, multiple of 16
#define TILE_CHUNKS (TILE_BYTES / 16) // 628 x b128 DMA chunks
#define NTILES  (NROWS / ROWS)        // 1664
#define NTHREADS 320                  // 10 wave32: 314 compute chunks -> 1 pass
#define KCHUNKS (DH / 4)              // 314 float4 chunks per row

typedef float v4f __attribute__((ext_vector_type(4)));

// CDNA5 async DMA: global -> LDS, 16B per lane, tracked by ASYNCcnt.
// GVS addressing: mem_addr = SADDR(64b sgpr pair) + VADDR(32b byte offset).
// VDST vgpr holds the workgroup-relative LDS byte address.
__device__ __forceinline__ void async_stage_tile(const float* __restrict__ x,
                                                 uint32_t lds_base_bytes,
                                                 int tile, int tid) {
    const uint32_t gbase = (uint32_t)tile * (uint32_t)TILE_BYTES;
#pragma unroll
    for (int c = 0; c < 2; ++c) {
        const int chunk = tid + c * NTHREADS;
        if (chunk < TILE_CHUNKS) {
            const uint32_t loff = lds_base_bytes + (uint32_t)chunk * 16u;
            const uint32_t goff = gbase + (uint32_t)chunk * 16u;
            asm volatile("global_load_async_to_lds_b128 %0, %1, %2"
                         :
                         : "v"(loff), "v"(goff), "s"(x)
                         : "memory");
        }
    }
}

__device__ __forceinline__ void wait_async_all() {
    asm volatile("s_wait_asynccnt 0x0" ::: "memory");
}

__global__ __launch_bounds__(NTHREADS)
void tabular_embedding_kernel(const float* __restrict__ x,
                              const float* __restrict__ cat_table,
                              const int*   __restrict__ cat_offsets,
                              const float* __restrict__ na_emb,
                              const float* __restrict__ w_cont,
                              const float* __restrict__ b_cont,
                              const float* __restrict__ pos_emb,
                              float*       __restrict__ out) {
    __shared__ float s_cat[VOCABT * H_];   // 360
    __shared__ float s_na[H_];             // 8
    __shared__ float s_w[NCONT * H_];      // 1168
    __shared__ float s_b[NCONT * H_];      // 1168
    __shared__ float s_pos[DH];            // 1256
    __shared__ int   s_off[NCAT];          // 11
    __shared__ __align__(16) float s_x[2][TILE_F]; // double-buffered x tiles

    const int tid = threadIdx.x;
    const int G   = gridDim.x;
    const int t0  = blockIdx.x;

    // Kick off the first tile's async DMA immediately (overlaps table load).
    if (t0 < NTILES) {
        async_stage_tile(x, (uint32_t)(uintptr_t)&s_x[0][0], t0, tid);
    }

    // Cooperative table staging into LDS (~3971 elements, once per block).
    for (int i = tid; i < VOCABT * H_; i += NTHREADS) s_cat[i] = cat_table[i];
    if (tid < H_)   s_na[tid]  = na_emb[tid];
    if (tid < NCAT) s_off[tid] = cat_offsets[tid];
    for (int i = tid; i < NCONT * H_; i += NTHREADS) {
        s_w[i] = w_cont[i];
        s_b[i] = b_cont[i];
    }
    for (int i = tid; i < DH; i += NTHREADS) s_pos[i] = pos_emb[i];

    // Per-thread, per-k-chunk invariants (independent of tile/row): hoist
    // them out of the tile loop entirely. Thread tid owns k = 4*tid..4*tid+3.
    const bool active = (tid < KCHUNKS);
    int   dx[4], he[4], off4[4];
    bool  cat4[4];
    float w4[4], bb4[4], pos4[4], na4[4];
    if (active) {
#pragma unroll
        for (int e = 0; e < 4; ++e) {
            const int k = 4 * tid + e;
            const int d = k % D_;          // emb_flat: d = k % 157
            const int h = k / D_;          //           h = k / 157
            dx[e]   = d;
            he[e]   = h;
            cat4[e] = (d < NCAT);
        }
    }

    int it = 0;
    for (int t = t0; t < NTILES; t += G, ++it) {
        const int buf = it & 1;

        // Complete this tile's DMA, then barrier: publishes async LDS writes
        // (and, on it==0, the table stores) to all waves, and guarantees the
        // other buffer is no longer being read before we overwrite it below.
        wait_async_all();
        __syncthreads();

        if (it == 0 && active) {
            // Table-dependent per-k constants (tables now visible).
#pragma unroll
            for (int e = 0; e < 4; ++e) {
                const int k = 4 * tid + e;
                pos4[e] = s_pos[k];
                na4[e]  = s_na[he[e]];
                if (cat4[e]) {
                    off4[e] = s_off[dx[e]];
                    w4[e] = 0.0f; bb4[e] = 0.0f;
                } else {
                    const int c = dx[e] - NCAT;
                    off4[e] = 0;
                    w4[e]  = s_w[c * H_ + he[e]];
                    bb4[e] = s_b[c * H_ + he[e]];
                }
            }
        }

        // Prefetch next tile into the other buffer while we compute.
        const int tn = t + G;
        if (tn < NTILES) {
            async_stage_tile(x, (uint32_t)(uintptr_t)&s_x[buf ^ 1][0], tn, tid);
        }

        if (active) {
            const float* sx = s_x[buf];
            const int r0 = t * ROWS;
            v4f* orow0 = (v4f*)(out + (size_t)r0 * DH) + tid;

#pragma unroll 4
            for (int rr = 0; rr < ROWS; ++rr) {
                const float* sxr = sx + rr * D_;
                v4f o;
#pragma unroll
                for (int e = 0; e < 4; ++e) {
                    const float xv = sxr[dx[e]];
                    float ev;
                    if (xv != xv) {                         // NaN -> na_emb
                        ev = na4[e];
                    } else if (cat4[e]) {                   // table gather
                        ev = s_cat[((int)xv + off4[e]) * H_ + he[e]];
                    } else {                                // x*W + b
                        ev = fmaf(xv, w4[e], bb4[e]);
                    }
                    const float v = ev + pos4[e];
                    // exact GELU: 0.5*v*(1+erf(v/sqrt(2)))
                    o[e] = 0.5f * v * (1.0f + erff(v * 0.70710678118654752440f));
                }
                __builtin_nontemporal_store(o, orow0 + (size_t)rr * (DH / 4));
            }
        }
        // Top-of-loop wait+barrier of the next iteration protects buf reuse.
    }
}

extern "C" void kernel_launch(void* const* d_in, const int* in_sizes, int n_in,
                              void* d_out, int out_size, void* d_ws, size_t ws_size,
                              hipStream_t stream) {
    (void)in_sizes; (void)n_in; (void)out_size; (void)d_ws; (void)ws_size;
    const float* x           = (const float*)d_in[0];
    const float* cat_table   = (const float*)d_in[1];
    const int*   cat_offsets = (const int*)  d_in[2];
    const float* na_emb      = (const float*)d_in[3];
    const float* w_cont      = (const float*)d_in[4];
    const float* b_cont      = (const float*)d_in[5];
    const float* pos_emb     = (const float*)d_in[6];
    float*       out         = (float*)d_out;

    // 1664 tiles; 832 blocks -> 2 tiles/block, 10 wave32 per block.
    const int grid = 832;
    tabular_embedding_kernel<<<grid, NTHREADS, 0, stream>>>(
        x, cat_table, cat_offsets, na_emb, w_cont, b_cont, pos_emb, out);
}